// SpectralKernelBank_34093450395915
// MI455X (gfx1250) — compile-verified
//
#include <hip/hip_runtime.h>
#include <hip/hip_bf16.h>
#include <math.h>

// ---------------------------------------------------------------------------
// SpectralKernelBank, folded:
//   out[t] = tanh( sum_k blend_w[k] * gate(t,k) * (x[t,:] . W_eff[k,:]) )
//   W_eff[k,h] = sum_f kw[k,f] * cos(2*pi*f*h/H)     (rfft real-part folded)
//   gate(t,k)  = (||x[t,:]||^2 > thr[k]^2)
// B=4, T=4096, H=2048, K=4, FB=1024. 16384 tokens total.
// ---------------------------------------------------------------------------

#define HDIM   2048
#define FBINS  1024
#define NKER   4
#define NTOK   16384           // B*T
#define NCHUNK (HDIM / 8)      // 256 h-chunks of 8 per 16-token tile
#define TILES  (NTOK / 16)     // 1024 wave-tiles
#define PFDIST 16              // prefetch distance in chunks (16B/lane each)

typedef __attribute__((ext_vector_type(2))) float v2f;
typedef __attribute__((ext_vector_type(8))) float v8f;

// ---------------------------------------------------------------------------
// Kernel 1: build the WMMA B-operand stream from kernel_weights.
// bstream[chunk][lane] = float4 laid out so kernel 2 consumes it verbatim:
//   col  = lane & 15 (only cols 0..3 carry W_eff, rest are zero N-columns)
//   half = lane >> 4 ; h0 = chunk*8 + half*4
//   .xy -> K-slots {0,1 | 2,3} of WMMA#0  = W_eff[col][h0], W_eff[col][h0+1]
//   .zw -> K-slots {0,1 | 2,3} of WMMA#1  = W_eff[col][h0+2], W_eff[col][h0+3]
// Phase is computed exactly mod 2048 so __cosf sees small arguments.
// ---------------------------------------------------------------------------
__global__ void sb_build_bstream(const float* __restrict__ kw,
                                 float4* __restrict__ bstream) {
    const int idx = blockIdx.x * blockDim.x + threadIdx.x;   // 0 .. 8191
    if (idx >= NCHUNK * 32) return;
    const int chunk = idx >> 5;
    const int lane  = idx & 31;
    const int col   = lane & 15;
    const int half  = lane >> 4;

    float4 v = make_float4(0.f, 0.f, 0.f, 0.f);
    if (col < NKER) {
        const int h0 = chunk * 8 + half * 4;
        const float* __restrict__ kwp = kw + col * FBINS;
        const float C = 6.283185307179586f / (float)HDIM;
        float s0 = 0.f, s1 = 0.f, s2 = 0.f, s3 = 0.f;
        for (int f = 0; f < FBINS; ++f) {
            const float w = kwp[f];
            s0 = __builtin_fmaf(w, __cosf((float)(((h0    ) * f) & (HDIM - 1)) * C), s0);
            s1 = __builtin_fmaf(w, __cosf((float)(((h0 + 1) * f) & (HDIM - 1)) * C), s1);
            s2 = __builtin_fmaf(w, __cosf((float)(((h0 + 2) * f) & (HDIM - 1)) * C), s2);
            s3 = __builtin_fmaf(w, __cosf((float)(((h0 + 3) * f) & (HDIM - 1)) * C), s3);
        }
        v = make_float4(s0, s1, s2, s3);
    }
    bstream[idx] = v;
}

// ---------------------------------------------------------------------------
// Kernel 2: one streaming pass over x. Each wave owns a 16-token tile.
// The 128KB B stream is staged once per workgroup into LDS with the CDNA5
// async Global->LDS path (global_load_async_to_lds_b128 / s_wait_asynccnt),
// then shared by all 8 waves. Per 8-wide H chunk: 1x global_load_b128 of x,
// 1x ds_load_b128 of B, 4 energy FMAs, 2x v_wmma_f32_16x16x4_f32.
// Epilogue: shuffle-reduce energy, gate, blend, tanh, 2x b128 store.
// ---------------------------------------------------------------------------
__global__ void __launch_bounds__(256)
sb_spectral_main(const float* __restrict__ x,
                 const float4* __restrict__ bstream,
                 const float* __restrict__ thr,
                 const float* __restrict__ bw,
                 float* __restrict__ out) {
    __shared__ float4 sB[NCHUNK * 32];               // 128 KB of WGP LDS

    const int lane = threadIdx.x & 31;
    const int wave = threadIdx.x >> 5;
    const int tile = (blockIdx.x * (blockDim.x >> 5) + wave) * 16;  // token base
    const int col  = lane & 15;      // WMMA N column (k) / token row for A
    const int half = lane >> 4;

    // ---- async-stage B stream into LDS: 256 threads x 32 entries x 16B ----
    {
        const int t = threadIdx.x;
        #pragma unroll
        for (int i = 0; i < (NCHUNK * 32) / 256; ++i) {
            const int e = i * 256 + t;
            const unsigned int  lds = (unsigned int)(uintptr_t)(&sB[e]);
            const unsigned long long ga = (unsigned long long)(uintptr_t)(bstream + e);
            asm volatile("global_load_async_to_lds_b128 %0, %1, off"
                         :: "v"(lds), "v"(ga) : "memory");
        }
        asm volatile("s_wait_asynccnt 0x0" ::: "memory");
        __syncthreads();
    }

    // A-operand pointer: lanes 0-15 stream h%8 in 0..3 of token (tile+col),
    // lanes 16-31 stream h%8 in 4..7 of the same token.
    const float4* __restrict__ xq =
        reinterpret_cast<const float4*>(x + (size_t)(tile + col) * HDIM) + half;

    v8f acc = {};            // 16x16 f32 accumulator (cols 0..3 meaningful)
    float esq_part = 0.f;    // partial sum of squares (energy^2)

    // rank-8 update step: x b128 load + LDS B fragment + 2 WMMAs
#define SB_STEP(c, DO_PF)                                                      \
    {                                                                          \
        const float4 xv = xq[2 * (c)];                                         \
        const float4 bv = sB[(c) * 32 + lane];                                 \
        if (DO_PF)                                                             \
            __builtin_prefetch((const void*)(xq + 2 * ((c) + PFDIST)), 0, 3);  \
        esq_part = __builtin_fmaf(xv.x, xv.x, esq_part);                       \
        esq_part = __builtin_fmaf(xv.y, xv.y, esq_part);                       \
        esq_part = __builtin_fmaf(xv.z, xv.z, esq_part);                       \
        esq_part = __builtin_fmaf(xv.w, xv.w, esq_part);                       \
        v2f a0 = {xv.x, xv.y}, a1 = {xv.z, xv.w};                              \
        v2f b0 = {bv.x, bv.y}, b1 = {bv.z, bv.w};                              \
        acc = __builtin_amdgcn_wmma_f32_16x16x4_f32(false, a0, false, b0,      \
                                                    (short)0, acc, false, false); \
        acc = __builtin_amdgcn_wmma_f32_16x16x4_f32(false, a1, false, b1,      \
                                                    (short)0, acc, false, false); \
    }

    #pragma unroll 4
    for (int c = 0; c < NCHUNK - PFDIST; ++c) SB_STEP(c, true)   // + prefetch
    #pragma unroll
    for (int c = NCHUNK - PFDIST; c < NCHUNK; ++c) SB_STEP(c, false)  // tail
#undef SB_STEP

    // energy^2 of token (lane&15): the two lane-halves covered disjoint h sets
    const float esq = esq_part + __shfl_xor(esq_part, 16, 32);

    // per-column (k) gate threshold^2 and blend weight; zero for padding cols
    const float tk   = (col < NKER) ? thr[col] : 0.f;
    const float thr2 = tk * tk;                        // |thr|^2
    const float bwk  = (col < NKER) ? bw[col] : 0.f;

    // acc[j] = scaled[token = half*8 + j][col]  (C/D layout)
    float outv[8];
    #pragma unroll
    for (int j = 0; j < 8; ++j) {
        const int tok_in_tile = half * 8 + j;
        const float e = __shfl(esq, tok_in_tile, 32);  // bpermute across lanes
        float s = (e > thr2) ? acc[j] * bwk : 0.f;     // gate * blend_w[k] * scaled
        s += __shfl_xor(s, 1, 32);                     // reduce k=0..3 group
        s += __shfl_xor(s, 2, 32);
        outv[j] = tanhf(s);
    }

    if (col == 0) {                                    // lanes 0 and 16 store
        float4* o = reinterpret_cast<float4*>(out + tile + half * 8);
        o[0] = make_float4(outv[0], outv[1], outv[2], outv[3]);
        o[1] = make_float4(outv[4], outv[5], outv[6], outv[7]);
    }
}

// ---------------------------------------------------------------------------
extern "C" void kernel_launch(void* const* d_in, const int* in_sizes, int n_in,
                              void* d_out, int out_size, void* d_ws, size_t ws_size,
                              hipStream_t stream) {
    (void)in_sizes; (void)n_in; (void)out_size; (void)ws_size;

    const float* x   = (const float*)d_in[0];   // (4,4096,2048) f32
    const float* kw  = (const float*)d_in[1];   // (4,1024) f32
    const float* thr = (const float*)d_in[2];   // (4,) f32
    const float* bw  = (const float*)d_in[3];   // (1,4) f32
    float* out = (float*)d_out;                 // (4,4096,1) f32

    float4* bstream = (float4*)d_ws;            // 256*32 float4 = 128 KB

    // 1) cosine-transform kernel_weights into WMMA B-operand stream
    sb_build_bstream<<<(NCHUNK * 32 + 255) / 256, 256, 0, stream>>>(kw, bstream);

    // 2) fused streaming pass: WMMA projection + energy gate + blend + tanh
    //    256 threads = 8 waves/block, 1 tile (16 tokens) per wave
    sb_spectral_main<<<TILES / 8, 256, 0, stream>>>(x, bstream, thr, bw, out);
}